// SDConv_35665408426627
// MI455X (gfx1250) — compile-verified
//
#include <hip/hip_runtime.h>

// SDConv as a GEMM: out[b,o,h,w] = sum_{c,k9} W[o, c*9+k9] * S_U[b, c*9+k9, hw]
// M = 128 (C_out), K = 1152 (C_in*9), N = 65536 (B*H*W), all f32.
// V_WMMA_F32_16X16X4_F32 (exact f32 semantics, matches reference dtype).
// Weights are pre-transposed to K-major [K][O] in d_ws so A-fragment loads
// are coalesced (64B per lane-half) instead of 16-line gathers.

typedef __attribute__((ext_vector_type(2))) float v2f;
typedef __attribute__((ext_vector_type(8))) float v8f;

#define C_IN   128
#define C_OUT  128
#define HDIM   128
#define WDIM   128
#define KW     (C_IN * 9)        // 1152
#define CCHUNK 16                // input channels staged per LDS chunk
#define KCHUNK (CCHUNK * 9)      // 144 K rows per chunk (/4 = 36 k-steps)
#define NPIX   64                // pixels per block (4 WMMA N-tiles)
// LDS: S stored as (k-pair) rows of interleaved {k_even,k_odd} per pixel.
// Row stride padded by 32 floats (128B) so the two lane halves (pair rows r,
// r+1) hit disjoint bank halves for ds_load_b64.
#define PSTRIDE (NPIX * 2 + 32)  // 160 floats per pair-row
#define NPAIRS  (KCHUNK / 2)     // 72

// ---- pre-pass: Wt [O][KW] -> Wk [KW][O] (K-major), 576 KB, L2-resident ----
__global__ __launch_bounds__(256)
void sdconv_transpose_w(const float* __restrict__ Wt, float* __restrict__ Wk) {
  const int i = blockIdx.x * 256 + threadIdx.x;   // i = k*C_OUT + o
  if (i < KW * C_OUT) {
    const int k = i >> 7;
    const int o = i & (C_OUT - 1);
    Wk[i] = Wt[o * KW + k];
  }
}

template <bool KMAJOR>
__global__ __launch_bounds__(256)
void sdconv_wmma_f32(const float* __restrict__ U, const float* __restrict__ V,
                     const float* __restrict__ Wt, float* __restrict__ Out) {
  __shared__ float sS[NPAIRS * PSTRIDE];   // 46080 B

  const int bid  = blockIdx.x;
  const int wseg = bid & 1;                 // which 64-pixel half of the row
  const int h    = (bid >> 1) & (HDIM - 1);
  const int bB   = bid >> 8;                // batch
  const int w0   = wseg * NPIX;

  const int tid    = threadIdx.x;
  const int lane   = tid & 31;
  const int wave   = tid >> 5;              // 0..7
  const int oBase  = wave * 16;             // 16 output channels per wave
  const int laneM  = lane & 15;
  const int laneHi = lane >> 4;             // 0|1: K(+2) / M(+8) half select

  const int p    = tid & (NPIX - 1);        // pixel within tile (phase-1 role)
  const int crow = tid >> 6;                // 0..3 (phase-1 channel stripe)

  v8f acc0 = {}; v8f acc1 = {}; v8f acc2 = {}; v8f acc3 = {};

  for (int cc = 0; cc < C_IN / CCHUNK; ++cc) {
    const int cBase = cc * CCHUNK;

    // ---- phase 1: materialize S_U chunk [KCHUNK x NPIX] into LDS ----
    #pragma unroll
    for (int i = 0; i < 4; ++i) {
      const int cl = crow + i * 4;                      // 0..15
      const int c  = cBase + cl;
      const int w  = w0 + p;
      const int rowBase = ((bB * C_IN + c) * HDIM + h) * WDIM;
      const float uc = U[rowBase + w];
      const float vc = V[rowBase + w];
      #pragma unroll
      for (int tap = 0; tap < 9; ++tap) {
        const int dy = tap / 3 - 1;
        const int dx = tap % 3 - 1;
        const int hh = h + dy;
        const int ww = w + dx;
        float un = 0.0f, vn = 0.0f;                     // zero-pad => dif = -center
        if (hh >= 0 && hh < HDIM && ww >= 0 && ww < WDIM) {
          const int nb = ((bB * C_IN + c) * HDIM + hh) * WDIM + ww;
          un = U[nb]; vn = V[nb];
        }
        const float dU = un - uc;
        const float dV = vn - vc;
        const float s  = dU * rsqrtf(1.0f + dV * dV);
        const int kidx = cl * 9 + tap;
        sS[(kidx >> 1) * PSTRIDE + p * 2 + (kidx & 1)] = s;  // pair-interleaved
      }
    }
    __syncthreads();

    // prefetch next chunk's U/V rows (global_prefetch_b8) so HBM fetch
    // overlaps this chunk's WMMA phase
    if (cc + 1 < C_IN / CCHUNK) {
      #pragma unroll
      for (int i = 0; i < 4; ++i) {
        const int c = cBase + CCHUNK + crow + i * 4;
        const int rowBase = ((bB * C_IN + c) * HDIM + h) * WDIM + w0 + p;
        __builtin_prefetch(&U[rowBase], 0, 0);
        __builtin_prefetch(&V[rowBase], 0, 0);
        if (h > 0)        { __builtin_prefetch(&U[rowBase - WDIM], 0, 0);
                            __builtin_prefetch(&V[rowBase - WDIM], 0, 0); }
        if (h < HDIM - 1) { __builtin_prefetch(&U[rowBase + WDIM], 0, 0);
                            __builtin_prefetch(&V[rowBase + WDIM], 0, 0); }
      }
    }

    // ---- phase 2: 36 K-steps of V_WMMA_F32_16X16X4_F32 ----
    const int kw0 = cBase * 9;   // chunk offset into full K
    #pragma unroll 4
    for (int ks = 0; ks < KCHUNK / 4; ++ks) {
      const int k0 = ks * 4;
      // A fragment (weights 16x4): lane<16 -> {K0,K1}, lane>=16 -> {K2,K3}, M=laneM
      v2f a;
      if (KMAJOR) {   // Wt is [KW][C_OUT]: coalesced 64B per lane-half
        const float* wp = Wt + (kw0 + k0 + laneHi * 2) * C_OUT + oBase + laneM;
        a.x = wp[0];
        a.y = wp[C_OUT];
      } else {        // fallback: [C_OUT][KW] gather
        const float* wp = Wt + (oBase + laneM) * KW + (kw0 + k0 + laneHi * 2);
        a.x = wp[0];
        a.y = wp[1];
      }
      // B fragments (S_U 4x16 per N-tile): V0={K0|K2}, V1={K1|K3}, N=laneM
      const int pr = (k0 >> 1) + laneHi;           // pair row
      const float* sp = &sS[pr * PSTRIDE + laneM * 2];
      v2f b0 = *(const v2f*)(sp +  0);
      v2f b1 = *(const v2f*)(sp + 32);
      v2f b2 = *(const v2f*)(sp + 64);
      v2f b3 = *(const v2f*)(sp + 96);
      acc0 = __builtin_amdgcn_wmma_f32_16x16x4_f32(false, a, false, b0, (short)0, acc0, false, false);
      acc1 = __builtin_amdgcn_wmma_f32_16x16x4_f32(false, a, false, b1, (short)0, acc1, false, false);
      acc2 = __builtin_amdgcn_wmma_f32_16x16x4_f32(false, a, false, b2, (short)0, acc2, false, false);
      acc3 = __builtin_amdgcn_wmma_f32_16x16x4_f32(false, a, false, b3, (short)0, acc3, false, false);
    }
    __syncthreads();
  }

  // ---- store D: VGPR i holds M=i (lanes 0-15) / M=i+8 (lanes 16-31), N=laneM ----
  #pragma unroll
  for (int i = 0; i < 8; ++i) {
    const int o = oBase + i + laneHi * 8;
    const int obase = ((bB * C_OUT + o) * HDIM + h) * WDIM + w0 + laneM;
    Out[obase +  0] = acc0[i];
    Out[obase + 16] = acc1[i];
    Out[obase + 32] = acc2[i];
    Out[obase + 48] = acc3[i];
  }
}

extern "C" void kernel_launch(void* const* d_in, const int* in_sizes, int n_in,
                              void* d_out, int out_size, void* d_ws, size_t ws_size,
                              hipStream_t stream) {
  (void)in_sizes; (void)n_in; (void)out_size;
  const float* U  = (const float*)d_in[0];
  const float* V  = (const float*)d_in[1];
  const float* Wt = (const float*)d_in[2];
  float* Out = (float*)d_out;

  // 4 batches * 128 rows * 2 half-rows = 1024 blocks, 256 threads (8 waves)
  dim3 grid(4 * HDIM * (WDIM / NPIX));
  dim3 block(256);

  const size_t wbytes = (size_t)KW * C_OUT * sizeof(float);   // 589824
  if (ws_size >= wbytes) {
    float* Wk = (float*)d_ws;
    sdconv_transpose_w<<<dim3((KW * C_OUT + 255) / 256), block, 0, stream>>>(Wt, Wk);
    sdconv_wmma_f32<true><<<grid, block, 0, stream>>>(U, V, Wk, Out);
  } else {
    sdconv_wmma_f32<false><<<grid, block, 0, stream>>>(U, V, Wt, Out);
  }
}